// YvDynamicMoELayer_81003083203658
// MI455X (gfx1250) — compile-verified
//
#include <hip/hip_runtime.h>
#include <hip/hip_bf16.h>

typedef __attribute__((ext_vector_type(2))) float v2f;
typedef __attribute__((ext_vector_type(8))) float v8f;

#define NTOK 8192
#define HDIM 4096
#define NEXP 64
#define CAP  160

// ---------------------------------------------------------------------------
// Gating GEMM: logits[NTOK][NEXP] = x[NTOK][HDIM] @ w[HDIM][NEXP]
// fp32 WMMA 16x16x4. Register-prefetch + ping-pong LDS double buffering.
// B staged k-pair-interleaved so each B fragment is one aligned ds_load_b64.
// Each workgroup: 64 rows x all 64 experts; 4 waves, wave w -> 16-row tile.
// ---------------------------------------------------------------------------
__global__ __launch_bounds__(128) void moe_gate_gemm(const float* __restrict__ x,
                                                     const float* __restrict__ w,
                                                     float* __restrict__ logits) {
    constexpr int SA  = 36;   // A row stride (floats): 16B-aligned, conflict-free frag reads
    constexpr int SBP = 160;  // B k-pair row stride (floats): mod 64 == 32 -> lane halves on disjoint banks
    __shared__ float As[2][64 * SA];    // 2 x 9216 B
    __shared__ float Bp[2][16 * SBP];   // 2 x 10240 B

    const int tid  = threadIdx.x;
    const int wave = tid >> 5;
    const int lane = tid & 31;
    const int half = lane >> 4;    // lane half selects K offset (A/B), row offset (C/D)
    const int lm   = lane & 15;
    const int r0   = blockIdx.x * 64;

    v8f acc[4];
#pragma unroll
    for (int n = 0; n < 4; ++n)
        acc[n] = (v8f){0.f, 0.f, 0.f, 0.f, 0.f, 0.f, 0.f, 0.f};

    float4 rA[4];
    float2 rB0[4], rB1[4];

    // ---- prefetch chunk 0 into registers, store into buffer 0 ----
#pragma unroll
    for (int it = 0; it < 4; ++it) {
        const int idx = tid + 128 * it;          // 0..511
        const int row = idx >> 3, c4 = idx & 7;  // A: 64 rows x 8 float4
        rA[it] = *reinterpret_cast<const float4*>(x + (size_t)(r0 + row) * HDIM + (c4 << 2));
        const int kp = idx >> 5, c2 = idx & 31;  // B: 16 kpairs x 32 col-pairs
        rB0[it] = *reinterpret_cast<const float2*>(w + (size_t)(2 * kp) * NEXP + 2 * c2);
        rB1[it] = *reinterpret_cast<const float2*>(w + (size_t)(2 * kp + 1) * NEXP + 2 * c2);
    }
#pragma unroll
    for (int it = 0; it < 4; ++it) {
        const int idx = tid + 128 * it;
        const int row = idx >> 3, c4 = idx & 7;
        *reinterpret_cast<float4*>(&As[0][row * SA + (c4 << 2)]) = rA[it];
        const int kp = idx >> 5, c2 = idx & 31;
        float4 bv = {rB0[it].x, rB1[it].x, rB0[it].y, rB1[it].y};  // interleave (k,k+1) per col
        *reinterpret_cast<float4*>(&Bp[0][kp * SBP + (c2 << 2)]) = bv;
    }
    __syncthreads();

    int cur = 0;
    for (int kc = 0; kc < HDIM; kc += 32) {
        const bool more = (kc + 32 < HDIM);
        // ---- issue next chunk's global loads early (overlap with WMMA) ----
        if (more) {
#pragma unroll
            for (int it = 0; it < 4; ++it) {
                const int idx = tid + 128 * it;
                const int row = idx >> 3, c4 = idx & 7;
                rA[it] = *reinterpret_cast<const float4*>(
                    x + (size_t)(r0 + row) * HDIM + (kc + 32) + (c4 << 2));
                const int kp = idx >> 5, c2 = idx & 31;
                rB0[it] = *reinterpret_cast<const float2*>(
                    w + (size_t)(kc + 32 + 2 * kp) * NEXP + 2 * c2);
                rB1[it] = *reinterpret_cast<const float2*>(
                    w + (size_t)(kc + 32 + 2 * kp + 1) * NEXP + 2 * c2);
            }
        }
        // ---- compute current buffer: 8 k-steps x 4 expert tiles ----
#pragma unroll
        for (int ks = 0; ks < 32; ks += 4) {
            // A fragment (16x4): lanes 0-15 K=ks,ks+1; lanes 16-31 K=ks+2,ks+3
            v2f a = *reinterpret_cast<const v2f*>(
                &As[cur][(wave * 16 + lm) * SA + ks + 2 * half]);
            const int kprow = (ks >> 1) + half;
#pragma unroll
            for (int n = 0; n < 4; ++n) {
                // B fragment pair (k,k+1 | k+2,k+3) contiguous -> single b64 load
                v2f b = *reinterpret_cast<const v2f*>(
                    &Bp[cur][kprow * SBP + ((n * 16 + lm) << 1)]);
                acc[n] = __builtin_amdgcn_wmma_f32_16x16x4_f32(
                    false, a, false, b, (short)0, acc[n], false, false);
            }
        }
        // ---- store prefetched regs into the other buffer ----
        if (more) {
#pragma unroll
            for (int it = 0; it < 4; ++it) {
                const int idx = tid + 128 * it;
                const int row = idx >> 3, c4 = idx & 7;
                *reinterpret_cast<float4*>(&As[1 - cur][row * SA + (c4 << 2)]) = rA[it];
                const int kp = idx >> 5, c2 = idx & 31;
                float4 bv = {rB0[it].x, rB1[it].x, rB0[it].y, rB1[it].y};
                *reinterpret_cast<float4*>(&Bp[1 - cur][kp * SBP + (c2 << 2)]) = bv;
            }
        }
        __syncthreads();
        cur ^= 1;
    }

    // C/D layout: lanes 0-15 -> M=i, N=lane; lanes 16-31 -> M=i+8, N=lane-16
#pragma unroll
    for (int n = 0; n < 4; ++n) {
#pragma unroll
        for (int i = 0; i < 8; ++i) {
            const int row = r0 + wave * 16 + i + 8 * half;
            const int col = n * 16 + lm;
            logits[(size_t)row * NEXP + col] = acc[n][i];
        }
    }
}

// ---------------------------------------------------------------------------
// Per-expert top-160: one workgroup per expert. Bitonic sort of 8192 packed
// (value,index) 64-bit keys in LDS. Ascending sort of (~monotone(f32))<<32|idx
// == value descending, index ascending (jax top_k tie-break). Converts the
// logits column in-place into the dispatch-mask column.
// ---------------------------------------------------------------------------
__global__ __launch_bounds__(1024) void moe_topk_scatter(float* __restrict__ logits_mask,
                                                         float* __restrict__ out_idx) {
    __shared__ unsigned long long keys[NTOK];   // 64 KB
    const int e   = blockIdx.x;
    const int tid = threadIdx.x;

    for (int i = tid; i < NTOK; i += 1024) {
        float v = logits_mask[(size_t)i * NEXP + e];
        unsigned u = __float_as_uint(v);
        unsigned m = (u & 0x80000000u) ? ~u : (u | 0x80000000u);  // ascending-orderable f32
        keys[i] = ((unsigned long long)(~m) << 32) | (unsigned)i;
        logits_mask[(size_t)i * NEXP + e] = 0.0f;  // zero this mask column
    }
    __threadfence();   // zeros visible before any 1.0 scatter below
    __syncthreads();

    for (int k = 2; k <= NTOK; k <<= 1) {
        for (int j = k >> 1; j > 0; j >>= 1) {
            for (int i = tid; i < NTOK; i += 1024) {
                int l = i ^ j;
                if (l > i) {
                    unsigned long long a = keys[i];
                    unsigned long long b = keys[l];
                    bool up = ((i & k) == 0);
                    if ((a > b) == up) { keys[i] = b; keys[l] = a; }
                }
            }
            __syncthreads();
        }
    }

    for (int c = tid; c < CAP; c += 1024) {
        unsigned token = (unsigned)(keys[c] & 0xffffffffu);
        out_idx[e * CAP + c] = (float)token;
        logits_mask[(size_t)token * NEXP + e] = 1.0f;
    }
}

// ---------------------------------------------------------------------------
// Load-balancing loss: expert_load = column sums of mask;
// loss = mean(load * log(load/mean(load) + 1e-8)).
// ---------------------------------------------------------------------------
__global__ __launch_bounds__(64) void moe_loss(const float* __restrict__ mask,
                                               float* __restrict__ out_loss) {
    __shared__ float loads[NEXP];
    const int e = threadIdx.x;  // 0..63
    float load = 0.f;
    for (int t = 0; t < NTOK; ++t)
        load += mask[(size_t)t * NEXP + e];
    loads[e] = load;
    __syncthreads();
    if (e == 0) {
        float total = 0.f;
        for (int i = 0; i < NEXP; ++i) total += loads[i];
        float mean = total / (float)NEXP;
        float acc = 0.f;
        for (int i = 0; i < NEXP; ++i)
            acc += loads[i] * logf(loads[i] / mean + 1e-8f);
        out_loss[0] = acc / (float)NEXP;
    }
}

// ---------------------------------------------------------------------------
extern "C" void kernel_launch(void* const* d_in, const int* in_sizes, int n_in,
                              void* d_out, int out_size, void* d_ws, size_t ws_size,
                              hipStream_t stream) {
    const float* x = (const float*)d_in[0];   // [8192, 4096] f32
    const float* w = (const float*)d_in[1];   // [4096, 64]   f32

    float* out     = (float*)d_out;
    float* out_idx = out;                               // [64, 160] (indices as float)
    float* mask    = out + NEXP * CAP;                  // [8192, 64] logits -> mask, in-place
    float* loss    = out + NEXP * CAP + NTOK * NEXP;    // scalar

    moe_gate_gemm<<<dim3(NTOK / 64), dim3(128), 0, stream>>>(x, w, mask);
    moe_topk_scatter<<<dim3(NEXP), dim3(1024), 0, stream>>>(mask, out_idx);
    moe_loss<<<dim3(1), dim3(64), 0, stream>>>(mask, loss);
}